// BasicModel_24000277250164
// MI455X (gfx1250) — compile-verified
//
#include <hip/hip_runtime.h>
#include <math.h>

typedef __attribute__((ext_vector_type(2))) float v2f;
typedef __attribute__((ext_vector_type(8))) float v8f;
typedef int v4i_ __attribute__((vector_size(16)));   // matches builtin param type

#define PHI_SIZE 720
#define T_SIZE   1024
#define NREC     256
#define BATCH    16
#define M_ROWS   (BATCH * PHI_SIZE)   // 11520 sinogram rows

#if __has_builtin(__builtin_amdgcn_global_load_async_to_lds_b128)
#define HAVE_ASYNC_LDS 1
#else
#define HAVE_ASYNC_LDS 0
#endif

__device__ __forceinline__ void async_wait_16() {
#if __has_builtin(__builtin_amdgcn_s_wait_asynccnt)
    __builtin_amdgcn_s_wait_asynccnt(16);
#else
    asm volatile("s_wait_asynccnt 16" ::: "memory");
#endif
}
__device__ __forceinline__ void async_wait_0() {
#if __has_builtin(__builtin_amdgcn_s_wait_asynccnt)
    __builtin_amdgcn_s_wait_asynccnt(0);
#else
    asm volatile("s_wait_asynccnt 0" ::: "memory");
#endif
}

// ---------------------------------------------------------------------------
// Kernel 1: h = irfft(kernel), 1024-point, direct inverse DFT.
// h[n] = (1/N) * ( Re K0 + (-1)^n Re K512 + 2*sum_{k=1}^{511} (Re Kk cos - Im Kk sin) )
// (pocketfft irfft ignores imag of bins 0 and N/2, matching this.)
// ---------------------------------------------------------------------------
__global__ __launch_bounds__(256) void build_h_kernel(const float* __restrict__ kre,
                                                      const float* __restrict__ kim,
                                                      float* __restrict__ h) {
    int n = blockIdx.x * 256 + threadIdx.x;
    if (n >= T_SIZE) return;
    float sum = 0.f;
    for (int k = 1; k < T_SIZE / 2; ++k) {
        int m = (k * n) & (T_SIZE - 1);           // exact phase reduction
        float ang = (float)m * (float)(2.0 * M_PI / (double)T_SIZE);
        float s, c;
        sincosf(ang, &s, &c);
        sum = fmaf(kre[k], c, sum);
        sum = fmaf(-kim[k], s, sum);
    }
    float edge = (n & 1) ? -kre[T_SIZE / 2] : kre[T_SIZE / 2];
    h[n] = (kre[0] + edge + 2.f * sum) * (1.f / (float)T_SIZE);
}

// ---------------------------------------------------------------------------
// Kernel 2: materialize the circulant B matrix, COLUMN-major:
//   Bc[n*1024 + s] = h[(n - s) mod 1024]
// ---------------------------------------------------------------------------
__global__ __launch_bounds__(256) void build_Bc_kernel(const float* __restrict__ h,
                                                       float* __restrict__ Bc) {
    int idx = blockIdx.x * 256 + threadIdx.x;   // 0 .. 1M-1
    int n = idx >> 10;
    int s = idx & (T_SIZE - 1);
    Bc[idx] = h[(n - s) & (T_SIZE - 1)];
}

// ---------------------------------------------------------------------------
// Kernel 3: circulant GEMM via V_WMMA_F32_16X16X4_F32.
//   filtered[r, t] = sum_s sinos[r, s] * h[(t-s) mod 1024]
// One wave -> 32(M) x 64(N) tile: 2 A-frags x 4 B-frags, K stepped by 4.
// Output scattered into [phi, T, B] layout for the backprojector.
// ---------------------------------------------------------------------------
__global__ __launch_bounds__(256) void conv_gemm_kernel(const float* __restrict__ A,
                                                        const float* __restrict__ Bc,
                                                        float* __restrict__ filt) {
    int tid  = threadIdx.x;
    int lane = tid & 31;
    int wv   = blockIdx.x * 8 + (tid >> 5);   // 0..5759
    int mTile = wv >> 4;                      // 0..359  (360 * 32 = 11520 rows)
    int nTile = wv & 15;                      // 0..15   (16 * 64 = 1024 cols)
    int m0 = mTile * 32;
    int n0 = nTile * 64;
    int half = lane >> 4;                     // 0: K{0,1}, 1: K{2,3}
    int mr   = lane & 15;

    const float* aptr0 = A  + (size_t)(m0 + mr) * T_SIZE + 2 * half;
    const float* aptr1 = aptr0 + (size_t)16 * T_SIZE;
    const float* bptr  = Bc + (size_t)(n0 + mr) * T_SIZE + 2 * half;

    v8f acc[2][4] = {};

    for (int kb = 0; kb < T_SIZE; kb += 4) {
        v2f af[2];
        af[0] = *(const v2f*)(aptr0 + kb);
        af[1] = *(const v2f*)(aptr1 + kb);
        v2f bf[4];
#pragma unroll
        for (int j = 0; j < 4; ++j)
            bf[j] = *(const v2f*)(bptr + (size_t)(16 * j) * T_SIZE + kb);
#pragma unroll
        for (int i = 0; i < 2; ++i)
#pragma unroll
            for (int j = 0; j < 4; ++j)
                acc[i][j] = __builtin_amdgcn_wmma_f32_16x16x4_f32(
                    /*neg_a=*/false, af[i], /*neg_b=*/false, bf[j],
                    /*c_mod=*/(short)0, acc[i][j],
                    /*reuse_a=*/false, /*reuse_b=*/false);
    }

    // C/D layout: lanes 0-15 -> M=v, N=lane ; lanes 16-31 -> M=8+v, N=lane-16.
#pragma unroll
    for (int i = 0; i < 2; ++i) {
        int rowBase = m0 + 16 * i + 8 * half;
#pragma unroll
        for (int j = 0; j < 4; ++j) {
            int n = n0 + 16 * j + mr;
#pragma unroll
            for (int v = 0; v < 8; ++v) {
                int row = rowBase + v;          // global sinogram row = b*720 + phi
                int bb  = row / PHI_SIZE;
                int ph  = row - bb * PHI_SIZE;
                filt[((size_t)ph * T_SIZE + n) * BATCH + bb] = acc[i][j][v];
            }
        }
    }
}

// ---------------------------------------------------------------------------
// Kernel 4: backprojection. filt layout [phi][T][B]; one angle slice = 64 KB.
// Double-buffered LDS staging with GLOBAL_LOAD_ASYNC_TO_LDS_B128 (ASYNCcnt):
// while interpolating slice a, the 16 async B128 loads/thread for slice a+1
// land in the other buffer. 128 KB LDS/WG -> 2 workgroups per 320 KB WGP.
// ---------------------------------------------------------------------------
__global__ __launch_bounds__(256) void backproject_kernel(const float* __restrict__ filt,
                                                          float* __restrict__ out) {
    __shared__ __align__(16) float sl[2][T_SIZE * BATCH];   // 2 x 64 KB

    int tid = threadIdx.x;
    int p   = blockIdx.x * 256 + tid;      // 0..65535
    int ix  = p >> 8;
    int iy  = p & 255;
    float X = ((float)ix + 0.5f) * (2.f / (float)NREC) - 1.f;
    float Y = ((float)iy + 0.5f) * (2.f / (float)NREC) - 1.f;

    const float INV_DT = (float)((double)T_SIZE / (2.0 * 1.4142135623730951)); // 1/DT
    float acc[BATCH] = {};

#if HAVE_ASYNC_LDS
    // Prefetch slice 0 into buffer 0 (16 async b128 per thread).
    {
        const char* g = (const char*)filt;
        char* l = (char*)&sl[0][0];
#pragma unroll
        for (int k = 0; k < 16; ++k) {
            int q = (tid + k * 256) * 16;
            __builtin_amdgcn_global_load_async_to_lds_b128(
                (__attribute__((address_space(1))) v4i_*)(g + q),
                (__attribute__((address_space(3))) v4i_*)(l + q), 0, 0);
        }
    }
#endif

    for (int a = 0; a < PHI_SIZE; ++a) {
        float* cur = &sl[a & 1][0];
#if HAVE_ASYNC_LDS
        int next = a + 1;
        if (next < PHI_SIZE) {
            const char* g = (const char*)(filt + (size_t)next * (T_SIZE * BATCH));
            char* l = (char*)&sl[next & 1][0];
#pragma unroll
            for (int k = 0; k < 16; ++k) {
                int q = (tid + k * 256) * 16;
                __builtin_amdgcn_global_load_async_to_lds_b128(
                    (__attribute__((address_space(1))) v4i_*)(g + q),
                    (__attribute__((address_space(3))) v4i_*)(l + q), 0, 0);
            }
            async_wait_16();   // in-order: current slice's 16 loads are done
        } else {
            async_wait_0();
        }
#else
        {
            const float4* s4 = (const float4*)(filt + (size_t)a * (T_SIZE * BATCH));
            float4* d4 = (float4*)cur;
            for (int q = tid; q < (T_SIZE * BATCH) / 4; q += 256)
                d4[q] = s4[q];
        }
#endif
        __syncthreads();

        float phi = ((float)a + 0.5f) * (float)(M_PI / (double)PHI_SIZE);
        float sn, cs;
        sincosf(phi, &sn, &cs);
        float t  = X * cs + Y * sn;
        float u  = fmaf(t, INV_DT, 511.5f);            // (t+rho)/dt - 0.5
        int   i0 = (int)floorf(u);
        i0 = min(max(i0, 0), T_SIZE - 2);
        float w  = u - (float)i0;
        float w0 = 1.f - w;

        const float* g0 = cur + i0 * BATCH;            // 16 floats, then next bin
#pragma unroll
        for (int b = 0; b < BATCH; ++b)
            acc[b] = fmaf(g0[b], w0, fmaf(g0[b + BATCH], w, acc[b]));
        __syncthreads();                               // protect buffer reuse
    }

    const float DPHI = (float)(M_PI / (double)PHI_SIZE);
#pragma unroll
    for (int b = 0; b < BATCH; ++b)
        out[(size_t)b * (NREC * NREC) + p] = acc[b] * DPHI;
}

// ---------------------------------------------------------------------------
extern "C" void kernel_launch(void* const* d_in, const int* in_sizes, int n_in,
                              void* d_out, int out_size, void* d_ws, size_t ws_size,
                              hipStream_t stream) {
    const float* sinos = (const float*)d_in[0];   // [16,720,1024] f32
    const float* kre   = (const float*)d_in[1];   // [513] f32
    const float* kim   = (const float*)d_in[2];   // [513] f32
    float* out = (float*)d_out;                   // [16,256,256] f32

    char*  ws   = (char*)d_ws;
    float* h    = (float*)(ws);                                        // 4 KB
    float* Bc   = (float*)(ws + 4096);                                 // 4 MB
    float* filt = (float*)(ws + 4096 + (size_t)T_SIZE * T_SIZE * 4);   // 45 MB, [phi][T][B]

    build_h_kernel <<<(T_SIZE + 255) / 256, 256, 0, stream>>>(kre, kim, h);
    build_Bc_kernel<<<(T_SIZE * T_SIZE) / 256, 256, 0, stream>>>(h, Bc);
    conv_gemm_kernel<<<PHI_SIZE, 256, 0, stream>>>(sinos, Bc, filt);
    backproject_kernel<<<(NREC * NREC) / 256, 256, 0, stream>>>(filt, out);
}